// Layer_1DPE_attention_86878598464036
// MI455X (gfx1250) — compile-verified
//
#include <hip/hip_runtime.h>

typedef float v2f __attribute__((ext_vector_type(2)));
typedef float v8f __attribute__((ext_vector_type(8)));

#define BATCH 8
#define CDIM 128
#define IDIM 128
#define NDIM 256
#define WSTR 130   // padded LDS row stride for weight strips (keeps b64 alignment, spreads banks)
#define KSTR 260   // padded LDS row stride for K/V strips

// Kernel 1: one block per (batch, 16-channel strip).
// Fused: 3 WMMA GEMMs (P1*A, P2*A, k*A) + rank-1-attention row-normalized
// reduction (Q cancels analytically) + residual + ReLU. Writes pre-BN
// activations to d_out and per-(block,channel) BN partials to ws.
__global__ __launch_bounds__(256)
void k1_gemm_attn(const float* __restrict__ A,
                  const float* __restrict__ P1,
                  const float* __restrict__ P2,
                  const float* __restrict__ kw,
                  float* __restrict__ out_pre,
                  float* __restrict__ psum,
                  float* __restrict__ psq)
{
    __shared__ float wl[3 * 16 * WSTR];   // weight strips: P1, P2, k
    __shared__ float Ks[16 * KSTR];       // K strip  [16 ch][256 n]
    __shared__ float Vs[16 * KSTR];       // V strip
    __shared__ float red0[16][17];
    __shared__ float red1[16][17];
    __shared__ float yv[16];
    __shared__ float bnS[16];
    __shared__ float bnQ[16];

    const int tid  = threadIdx.x;
    const int lane = tid & 31;
    const int wave = tid >> 5;
    const int bx   = blockIdx.x;
    const int b    = bx >> 3;   // batch index
    const int ot   = bx & 7;    // 16-channel strip index

    // ---- warm L2/L0 for this block's 128KB A[b] slice while we stage weights:
    // one byte per 128B line -> global_prefetch_b8 (no LOADcnt cost, ISA 10.5)
    const float* Ab = A + b * IDIM * NDIM;   // A[b][i][n]
    {
        const char* pbase = (const char*)Ab;
        #pragma unroll
        for (int p = 0; p < 4; ++p)
            __builtin_prefetch(pbase + (tid + p * 256) * 128, 0, 3);
    }

    // ---- stage weight strips (16 rows x 128 k) into padded LDS ----
    for (int m = 0; m < 3; ++m) {
        const float* src = (m == 0) ? P1 : (m == 1) ? P2 : kw;
        for (int idx = tid; idx < 16 * IDIM; idx += 256) {
            const int o = idx >> 7, i = idx & 127;
            wl[m * 16 * WSTR + o * WSTR + i] = src[(ot * 16 + o) * IDIM + i];
        }
    }
    __syncthreads();

    // ---- WMMA phase: each wave owns two 16x16 n-tiles, K swept in steps of 4 ----
    const int nl     = lane & 15;               // A-frag: M row; B-frag: N col
    const int hi     = lane >> 4;
    const int koff   = hi * 2;                  // K split across lane halves

    v8f accA1[2];
    for (int t = 0; t < 2; ++t) {
        const int ntile = wave * 2 + t;
        const int n     = ntile * 16 + nl;

        v8f aA1 = {0.f,0.f,0.f,0.f,0.f,0.f,0.f,0.f};
        v8f aV  = {0.f,0.f,0.f,0.f,0.f,0.f,0.f,0.f};
        v8f aK  = {0.f,0.f,0.f,0.f,0.f,0.f,0.f,0.f};

        #pragma unroll 4
        for (int kb = 0; kb < IDIM; kb += 4) {
            v2f bf;                                   // B-frag: A[i..i+1][n]
            bf.x = Ab[(kb + koff) * NDIM + n];
            bf.y = Ab[(kb + koff + 1) * NDIM + n];
            const int wb = nl * WSTR + kb + koff;     // A-frag: W[o][i..i+1]
            const v2f f1 = *reinterpret_cast<const v2f*>(&wl[0 * 16 * WSTR + wb]);
            const v2f f2 = *reinterpret_cast<const v2f*>(&wl[1 * 16 * WSTR + wb]);
            const v2f fk = *reinterpret_cast<const v2f*>(&wl[2 * 16 * WSTR + wb]);
            aA1 = __builtin_amdgcn_wmma_f32_16x16x4_f32(false, f1, false, bf, (short)0, aA1, false, false);
            aV  = __builtin_amdgcn_wmma_f32_16x16x4_f32(false, f2, false, bf, (short)0, aV,  false, false);
            aK  = __builtin_amdgcn_wmma_f32_16x16x4_f32(false, fk, false, bf, (short)0, aK,  false, false);
        }

        #pragma unroll
        for (int j = 0; j < 8; ++j) {                 // D layout: M = j + 8*hi, N = nl
            const int ol = j + (hi << 3);
            Ks[ol * KSTR + n] = aK[j];
            Vs[ol * KSTR + n] = aV[j];
        }
        accA1[t] = aA1;
    }
    __syncthreads();

    // ---- per-channel reductions: ||K||_2 and sum(|K|*V) over n ----
    {
        const int c = tid >> 4, ch = tid & 15;
        float sq = 0.f, sd = 0.f;
        #pragma unroll
        for (int u = 0; u < 16; ++u) {
            const float kv = Ks[c * KSTR + ch * 16 + u];
            const float vv = Vs[c * KSTR + ch * 16 + u];
            sq += kv * kv;
            sd += fabsf(kv) * vv;
        }
        red0[c][ch] = sq;
        red1[c][ch] = sd;
    }
    __syncthreads();
    if (tid < 16) {
        float sq = 0.f, sd = 0.f;
        #pragma unroll
        for (int u = 0; u < 16; ++u) { sq += red0[tid][u]; sd += red1[tid][u]; }
        yv[tid]  = 0.1f * sd / sqrtf(sq);   // Y is a per-(b,c) scalar, pre-scaled by 0.1
        bnS[tid] = 0.f;
        bnQ[tid] = 0.f;
    }
    __syncthreads();

    // ---- residual + ReLU, write pre-BN output, accumulate BN partials ----
    float ps[8], pq[8];
    #pragma unroll
    for (int j = 0; j < 8; ++j) { ps[j] = 0.f; pq[j] = 0.f; }

    for (int t = 0; t < 2; ++t) {
        const int n = (wave * 2 + t) * 16 + nl;
        #pragma unroll
        for (int j = 0; j < 8; ++j) {
            const int ol = j + (hi << 3);
            float v = accA1[t][j] + Vs[ol * KSTR + n] + yv[ol];
            v = fmaxf(v, 0.f);
            out_pre[(b * CDIM + ot * 16 + ol) * NDIM + n] = v;
            ps[j] += v;
            pq[j] += v * v;
        }
    }
    #pragma unroll
    for (int j = 0; j < 8; ++j) {
        atomicAdd(&bnS[j + (hi << 3)], ps[j]);
        atomicAdd(&bnQ[j + (hi << 3)], pq[j]);
    }
    __syncthreads();
    if (tid < 16) {
        psum[bx * 16 + tid] = bnS[tid];
        psq [bx * 16 + tid] = bnQ[tid];
    }
}

// Kernel 2: fold batch partials into fused BN scale/shift per channel.
__global__ __launch_bounds__(128)
void k2_bn_stats(const float* __restrict__ psum,
                 const float* __restrict__ psq,
                 const float* __restrict__ gamma,
                 const float* __restrict__ beta,
                 float* __restrict__ scale,
                 float* __restrict__ shift)
{
    const int c  = threadIdx.x;        // 0..127
    const int ot = c >> 4, cl = c & 15;
    float s = 0.f, q = 0.f;
    #pragma unroll
    for (int bb = 0; bb < BATCH; ++bb) {
        s += psum[((bb << 3) + ot) * 16 + cl];
        q += psq [((bb << 3) + ot) * 16 + cl];
    }
    const float inv  = 1.f / (float)(BATCH * NDIM);
    const float mean = s * inv;
    const float var  = q * inv - mean * mean;
    const float rstd = rsqrtf(var + 1e-5f);
    const float sc   = gamma[c] * rstd;
    scale[c] = sc;
    shift[c] = beta[c] - mean * sc;
}

// Kernel 3: in-place BatchNorm apply.
__global__ __launch_bounds__(256)
void k3_bn_apply(float* __restrict__ out,
                 const float* __restrict__ scale,
                 const float* __restrict__ shift)
{
    const int idx = blockIdx.x * 256 + threadIdx.x;
    const int c   = (idx >> 8) & (CDIM - 1);   // NDIM == 256
    out[idx] = out[idx] * scale[c] + shift[c];
}

extern "C" void kernel_launch(void* const* d_in, const int* in_sizes, int n_in,
                              void* d_out, int out_size, void* d_ws, size_t ws_size,
                              hipStream_t stream)
{
    const float* A     = (const float*)d_in[0];
    const float* P1    = (const float*)d_in[1];
    const float* P2    = (const float*)d_in[2];
    // d_in[3] = q: unused — |Q_i| cancels out of the row-normalized rank-1 attention.
    const float* kw    = (const float*)d_in[4];
    const float* gamma = (const float*)d_in[5];
    const float* beta  = (const float*)d_in[6];

    float* out = (float*)d_out;
    float* ws  = (float*)d_ws;
    float* psum  = ws;          // 64 blocks * 16 ch
    float* psq   = ws + 1024;
    float* scale = ws + 2048;   // 128
    float* shift = ws + 2176;   // 128

    k1_gemm_attn<<<BATCH * 8, 256, 0, stream>>>(A, P1, P2, kw, out, psum, psq);
    k2_bn_stats<<<1, 128, 0, stream>>>(psum, psq, gamma, beta, scale, shift);
    k3_bn_apply<<<(BATCH * CDIM * NDIM) / 256, 256, 0, stream>>>(out, scale, shift);
}